// Attention_3599182594731
// MI455X (gfx1250) — compile-verified
//
#include <hip/hip_runtime.h>

// ---------------------------------------------------------------------------
// MI455X (gfx1250, wave32) implementation of transposed channel attention.
// Heavy GEMMs use V_WMMA_F32_16X16X4_F32 (fp32 in/out -> exact precision vs
// fp32 reference; problem is HBM-bound, ~3 GB @ 23.3 TB/s, so fp32 WMMA rate
// is not the limiter). Tile staging uses GLOBAL_LOAD_ASYNC_TO_LDS_B128
// (ASYNCcnt-tracked direct global->LDS copies) when the toolchain exposes it.
// ---------------------------------------------------------------------------

typedef __attribute__((ext_vector_type(2))) float v2f;
typedef __attribute__((ext_vector_type(8))) float v8f;
typedef __attribute__((ext_vector_type(4))) int v4i;

#if __has_builtin(__builtin_amdgcn_wmma_f32_16x16x4_f32)
#define HAS_WMMA_F32 1
#else
#define HAS_WMMA_F32 0
#endif

#if __has_builtin(__builtin_amdgcn_global_load_async_to_lds_b128)
#define HAS_ASYNC_LDS 1
#else
#define HAS_ASYNC_LDS 0
#endif

typedef __attribute__((address_space(1))) v4i GV4;  // global v4i
typedef __attribute__((address_space(3))) v4i LV4;  // LDS v4i

__device__ __forceinline__ v8f wmma4_f32(v2f a, v2f b, v8f c) {
#if HAS_WMMA_F32
  // 8 args: (neg_a, A, neg_b, B, c_mod, C, reuse_a, reuse_b)
  return __builtin_amdgcn_wmma_f32_16x16x4_f32(false, a, false, b, (short)0, c,
                                               false, false);
#else
  (void)a; (void)b;
  return c;  // placeholder so compile survives; histogram will show wmma==0
#endif
}

// Direct global -> LDS async copy of 16 bytes (falls back to VGPR round-trip).
__device__ __forceinline__ void async_g2l_b128(float* lds, const float* g) {
#if HAS_ASYNC_LDS
  __builtin_amdgcn_global_load_async_to_lds_b128((GV4*)(v4i*)(float*)g,
                                                 (LV4*)(v4i*)lds, 0, 0);
#else
  *(float4*)lds = *(const float4*)g;
#endif
}

__device__ __forceinline__ void async_join() {
#if HAS_ASYNC_LDS
#if __has_builtin(__builtin_amdgcn_s_wait_asynccnt)
  __builtin_amdgcn_s_wait_asynccnt(0);
#else
  asm volatile("s_wait_asynccnt 0x0" ::: "memory");
#endif
#endif
}

// ---------------------------------------------------------------------------
// GEMM: C[b] = A[b] (MxK, row-major lda) * B[b] (KxN, row-major ldb)
// M = gridDim.y*64, N = gridDim.x*64, K multiple of 64.
// Block: 256 threads = 8 waves; wave (wm,wn) computes a 16x32 output tile.
// LDS staging with pad strides 68/80 -> conflict-free fragment gathers.
// ---------------------------------------------------------------------------
#define KC 64
#define LDA_SH 68
#define LDB_SH 80

__global__ __launch_bounds__(256) void gemm_wmma_f32(
    const float* __restrict__ A, long aBatch, int lda,
    const float* __restrict__ Bm, long bBatch, int ldb,
    float* __restrict__ C, long cBatch, int ldc, int K) {
  __shared__ float lA[64 * LDA_SH];
  __shared__ float lB[KC * LDB_SH];

  const int tid = threadIdx.x;
  const int lane = tid & 31, wave = tid >> 5;
  const int lo = lane & 15, hi = lane >> 4;
  const int wm = wave >> 1, wn = wave & 1;
  const int bz = blockIdx.z;
  const int m0 = blockIdx.y * 64;
  const long n0 = (long)blockIdx.x * 64;

  const float* gA = A + (long)bz * aBatch;
  const float* gB = Bm + (long)bz * bBatch;
  float* gC = C + (long)bz * cBatch;

  v8f acc0 = {};
  v8f acc1 = {};

  for (int kc = 0; kc < K; kc += KC) {
    __syncthreads();
#pragma unroll
    for (int i = 0; i < 4; ++i) {  // stage A tile: 64 rows x 64 k
      int idx = tid + 256 * i;     // 0..1023 float4s
      int row = idx >> 4, col4 = (idx & 15) << 2;
      async_g2l_b128(&lA[row * LDA_SH + col4],
                     gA + (long)(m0 + row) * lda + kc + col4);
    }
#pragma unroll
    for (int i = 0; i < 4; ++i) {  // stage B tile: 64 k x 64 n
      int idx = tid + 256 * i;
      int row = idx >> 4, col4 = (idx & 15) << 2;
      async_g2l_b128(&lB[row * LDB_SH + col4],
                     gB + (long)(kc + row) * ldb + n0 + col4);
    }
    if (kc + KC < K) {  // hint next B chunk toward the caches
      __builtin_prefetch(gB + (long)(kc + KC) * ldb + n0 + (tid & 15) * 4, 0, 1);
    }
    async_join();
    __syncthreads();
#pragma unroll
    for (int k = 0; k < KC; k += 4) {
      // A fragment: lane lo -> M row, hi selects K pair -> single b64 read
      v2f a = *(const v2f*)(&lA[(wm * 16 + lo) * LDA_SH + k + 2 * hi]);
      // B fragments for the two 16-wide N sub-tiles
      int r0 = (k + 2 * hi) * LDB_SH + wn * 32 + lo;
      v2f b0, b1;
      b0.x = lB[r0];
      b0.y = lB[r0 + LDB_SH];
      b1.x = lB[r0 + 16];
      b1.y = lB[r0 + 16 + LDB_SH];
      acc0 = wmma4_f32(a, b0, acc0);
      acc1 = wmma4_f32(a, b1, acc1);
    }
  }

#pragma unroll
  for (int r = 0; r < 8; ++r) {  // C/D layout: M = r + 8*hi, N = lo
    long o = (long)(m0 + wm * 16 + r + 8 * hi);
    long n = n0 + wn * 32 + lo;
    gC[o * (long)ldc + n] = acc0[r];
    gC[o * (long)ldc + n + 16] = acc1[r];
  }
}

// ---------------------------------------------------------------------------
// Depthwise 3x3, padding=1 (cross-correlation, matches lax/torch conv).
// 4 outputs per thread: per row 1 aligned float4 + 2 edge scalars.
// ---------------------------------------------------------------------------
__global__ __launch_bounds__(256) void dwconv3x3(const float* __restrict__ in,
                                                 const float* __restrict__ w,
                                                 float* __restrict__ out) {
  const int c = blockIdx.y;  // 0..575
  const int b = blockIdx.z;
  const int idx = blockIdx.x * 256 + threadIdx.x;  // 0..16383 (pixel quads)
  const int y = idx >> 6, x4 = (idx & 63) << 2;
  const float* p = in + ((long)b * 576 + c) * 65536;
  const float* wc = w + c * 9;
  float s0 = 0.f, s1 = 0.f, s2 = 0.f, s3 = 0.f;
#pragma unroll
  for (int j = 0; j < 3; ++j) {
    int yy = y + j - 1;
    if (yy < 0 || yy > 255) continue;
    const float* row = p + yy * 256;
    float4 m = *(const float4*)(row + x4);
    float lft = (x4 > 0) ? row[x4 - 1] : 0.f;
    float rgt = (x4 < 252) ? row[x4 + 4] : 0.f;
    float w0 = wc[j * 3 + 0], w1 = wc[j * 3 + 1], w2 = wc[j * 3 + 2];
    s0 += w0 * lft + w1 * m.x + w2 * m.y;
    s1 += w0 * m.x + w1 * m.y + w2 * m.z;
    s2 += w0 * m.y + w1 * m.z + w2 * m.w;
    s3 += w0 * m.z + w1 * m.w + w2 * rgt;
  }
  float4 r;
  r.x = s0; r.y = s1; r.z = s2; r.w = s3;
  *(float4*)(out + ((long)b * 576 + c) * 65536 + y * 256 + x4) = r;
}

// ---------------------------------------------------------------------------
// Per-channel L2 norm over HW for q (ch 0..191) and k (ch 192..383).
// ---------------------------------------------------------------------------
__global__ __launch_bounds__(256) void l2norm(const float* __restrict__ dwq,
                                              float* __restrict__ norms) {
  const int ch = blockIdx.x;  // 0..383
  const int b = blockIdx.y;
  const float4* p = (const float4*)(dwq + ((long)b * 576 + ch) * 65536);
  float s = 0.f;
  for (int i = threadIdx.x; i < 16384; i += 256) {
    float4 v = p[i];
    s += v.x * v.x + v.y * v.y + v.z * v.z + v.w * v.w;
  }
  __shared__ float red[256];
  red[threadIdx.x] = s;
  __syncthreads();
  for (int st = 128; st > 0; st >>= 1) {
    if (threadIdx.x < st) red[threadIdx.x] += red[threadIdx.x + st];
    __syncthreads();
  }
  if (threadIdx.x == 0) norms[b * 384 + ch] = sqrtf(red[0]);
}

// ---------------------------------------------------------------------------
// Gram: S[b,h][32x32 padded] = q_raw @ k_raw^T  (24x24 live region).
// Split-K over gridDim.y blocks; WMMA over LDS-staged 32x128 tiles
// (rows 24..31 zeroed once); fp32 atomicAdd to combine partials.
// ---------------------------------------------------------------------------
#define GLD 132
__global__ __launch_bounds__(128) void gram_wmma(const float* __restrict__ dwq,
                                                 float* __restrict__ S) {
  __shared__ float lQ[32 * GLD];
  __shared__ float lK[32 * GLD];
  const int tid = threadIdx.x;
  const int lane = tid & 31, wave = tid >> 5;
  const int lo = lane & 15, hi = lane >> 4;
  const int bh = blockIdx.x;  // 0..31
  const int b = bh >> 3, h = bh & 7;
  const long nBase = (long)blockIdx.y * 1024;
  const float* q = dwq + ((long)b * 576 + h * 24) * 65536;
  const float* kp = dwq + ((long)b * 576 + 192 + h * 24) * 65536;

  for (int i = tid; i < 8 * GLD; i += 128) {  // zero pad rows 24..31
    lQ[24 * GLD + i] = 0.f;
    lK[24 * GLD + i] = 0.f;
  }

  v8f a00 = {}, a01 = {}, a10 = {}, a11 = {};

  for (int it = 0; it < 8; ++it) {
    long nt = nBase + it * 128;
    __syncthreads();
#pragma unroll
    for (int i = 0; i < 6; ++i) {  // stage 24 rows x 128 n (coalesced)
      int idx = tid + 128 * i;     // 0..767 float4s
      int row = idx >> 5, col4 = (idx & 31) << 2;
      async_g2l_b128(&lQ[row * GLD + col4],
                     q + (long)row * 65536 + nt + col4);
      async_g2l_b128(&lK[row * GLD + col4],
                     kp + (long)row * 65536 + nt + col4);
    }
    async_join();
    __syncthreads();
    const int nb = wave * 32;  // each wave reduces a 32-wide n slice
#pragma unroll
    for (int k = 0; k < 32; k += 4) {
      int off = nb + k + 2 * hi;
      v2f qa0 = *(const v2f*)(&lQ[lo * GLD + off]);
      v2f qa1 = *(const v2f*)(&lQ[(16 + lo) * GLD + off]);
      v2f kb0 = *(const v2f*)(&lK[lo * GLD + off]);
      v2f kb1 = *(const v2f*)(&lK[(16 + lo) * GLD + off]);
      a00 = wmma4_f32(qa0, kb0, a00);
      a01 = wmma4_f32(qa0, kb1, a01);
      a10 = wmma4_f32(qa1, kb0, a10);
      a11 = wmma4_f32(qa1, kb1, a11);
    }
  }

  float* Sp = S + bh * 1024;
#pragma unroll
  for (int r = 0; r < 8; ++r) {
    int m = r + 8 * hi;
    atomicAdd(&Sp[m * 32 + lo], a00[r]);
    atomicAdd(&Sp[m * 32 + 16 + lo], a01[r]);
    atomicAdd(&Sp[(16 + m) * 32 + lo], a10[r]);
    atomicAdd(&Sp[(16 + m) * 32 + 16 + lo], a11[r]);
  }
}

// ---------------------------------------------------------------------------
// Per batch: normalize Gram -> *temperature -> softmax rows -> compose
// Mc[o][h*24+e] = sum_d P[o][h*24+d] * attn[h][d][e]   (192x192)
// so the final pass is a single GEMM over v.
// ---------------------------------------------------------------------------
__global__ __launch_bounds__(256) void softmax_compose(
    const float* __restrict__ S, const float* __restrict__ norms,
    const float* __restrict__ temp, const float* __restrict__ P,
    float* __restrict__ Mc) {
  const int b = blockIdx.x;
  const int tid = threadIdx.x;
  __shared__ float attn[192][25];

  if (tid < 192) {
    int h = tid / 24, d = tid % 24;
    const float* Sp = S + ((long)(b * 8 + h)) * 1024 + d * 32;
    float nq = fmaxf(norms[b * 384 + h * 24 + d], 1e-12f);
    float t = temp[h];
    float vals[24];
    float mx = -1e30f;
#pragma unroll
    for (int e = 0; e < 24; ++e) {
      float nk = fmaxf(norms[b * 384 + 192 + h * 24 + e], 1e-12f);
      float v = Sp[e] * t / (nq * nk);
      vals[e] = v;
      mx = fmaxf(mx, v);
    }
    float sum = 0.f;
#pragma unroll
    for (int e = 0; e < 24; ++e) {
      float ev = expf(vals[e] - mx);
      vals[e] = ev;
      sum += ev;
    }
    float inv = 1.f / sum;
#pragma unroll
    for (int e = 0; e < 24; ++e) attn[tid][e] = vals[e] * inv;
  }
  __syncthreads();

  for (int idx = tid; idx < 192 * 192; idx += 256) {
    int o = idx / 192, c = idx % 192;
    int h = c / 24, e = c % 24;
    const float* Pp = P + o * 192 + h * 24;
    float s = 0.f;
#pragma unroll
    for (int d = 0; d < 24; ++d) s += Pp[d] * attn[h * 24 + d][e];
    Mc[(long)b * 36864 + idx] = s;
  }
}

// ---------------------------------------------------------------------------
extern "C" void kernel_launch(void* const* d_in, const int* in_sizes, int n_in,
                              void* d_out, int out_size, void* d_ws,
                              size_t ws_size, hipStream_t stream) {
  (void)in_sizes; (void)n_in; (void)out_size; (void)ws_size;
  const float* x = (const float*)d_in[0];       // (4,192,256,256)
  const float* qkv_w = (const float*)d_in[1];   // (576,192)
  const float* dw_w = (const float*)d_in[2];    // (576,1,3,3)
  const float* proj_w = (const float*)d_in[3];  // (192,192)
  const float* temp = (const float*)d_in[4];    // (8,1,1)
  float* out = (float*)d_out;

  const long HW = 65536;
  float* qkv = (float*)d_ws;                  // 4*576*HW
  float* dwq = qkv + (long)4 * 576 * HW;      // 4*576*HW
  float* norms = dwq + (long)4 * 576 * HW;    // 4*384
  float* S = norms + 4 * 384;                 // 4*8*32*32
  float* Mc = S + 32768;                      // 4*192*192

  // 1) qkv = Wqkv @ x  (per pixel): M=576, N=65536, K=192, A shared.
  gemm_wmma_f32<<<dim3(1024, 9, 4), 256, 0, stream>>>(
      qkv_w, 0L, 192, x, 192 * HW, (int)HW, qkv, 576 * HW, (int)HW, 192);
  // 2) depthwise 3x3
  dwconv3x3<<<dim3(64, 576, 4), 256, 0, stream>>>(qkv, dw_w, dwq);
  // 3) L2 norms for q,k channels
  l2norm<<<dim3(384, 4), 256, 0, stream>>>(dwq, norms);
  // 4) Gram matrices (split-K + atomics)
  (void)hipMemsetAsync(S, 0, 32768 * sizeof(float), stream);
  gram_wmma<<<dim3(32, 64), 128, 0, stream>>>(dwq, S);
  // 5) softmax + fold proj into a per-batch 192x192 matrix
  softmax_compose<<<4, 256, 0, stream>>>(S, norms, temp, proj_w, Mc);
  // 6) y = Mc @ v : M=192, N=65536, K=192, per-batch A.
  gemm_wmma_f32<<<dim3(1024, 3, 4), 256, 0, stream>>>(
      Mc, 192L * 192, 192, dwq + 384 * HW, 576 * HW, (int)HW, out, 192 * HW,
      (int)HW, 192);
}